// NoDeviceHeteroGAT1Layer_60318520705355
// MI455X (gfx1250) — compile-verified
//
#include <hip/hip_runtime.h>
#include <hip/hip_bf16.h>
#include <math.h>

// Problem constants (match reference)
#define NTASK 100000
#define NDATA 50000
#define N_EDT 2000000
#define N_ETT 1000000

typedef __attribute__((ext_vector_type(2))) float v2f;
typedef __attribute__((ext_vector_type(8))) float v8f;

__device__ __forceinline__ void atomic_max_float(float* addr, float val) {
    // order-preserving int trick: works for mixed signs with -inf init
    if (val >= 0.0f) atomicMax((int*)addr, __float_as_int(val));
    else             atomicMin((unsigned int*)addr, __float_as_uint(val));
}

// ---------------------------------------------------------------------------
// Pack weights: Wcat[12][256] = [Wr_dt | Wl_tt | Wr_tt | Wres_dt | Wres_tt],
// bias_cat[256], Wd[8][64] = Wl_dt zero-padded, bias_d[64].
// ---------------------------------------------------------------------------
__global__ void prep_kernel(const float* __restrict__ Wl_dt, const float* __restrict__ bl_dt,
                            const float* __restrict__ Wr_dt, const float* __restrict__ br_dt,
                            const float* __restrict__ Wres_dt, const float* __restrict__ bias_dt,
                            const float* __restrict__ Wl_tt, const float* __restrict__ bl_tt,
                            const float* __restrict__ Wr_tt, const float* __restrict__ br_tt,
                            const float* __restrict__ Wres_tt, const float* __restrict__ bias_tt,
                            float* __restrict__ Wcat, float* __restrict__ bias_cat,
                            float* __restrict__ Wd, float* __restrict__ bias_d) {
    const int tid = blockIdx.x * blockDim.x + threadIdx.x;
    const int nthr = blockDim.x * gridDim.x;
    for (int i = tid; i < 12 * 256; i += nthr) {
        const int r = i >> 8, c = i & 255;
        float v;
        if      (c < 64)  v = Wr_dt[r * 64 + c];
        else if (c < 128) v = Wl_tt[r * 64 + (c - 64)];
        else if (c < 192) v = Wr_tt[r * 64 + (c - 128)];
        else if (c < 224) v = Wres_dt[r * 32 + (c - 192)];
        else              v = Wres_tt[r * 32 + (c - 224)];
        Wcat[i] = v;
    }
    for (int c = tid; c < 256; c += nthr) {
        float v;
        if      (c < 64)  v = br_dt[c];
        else if (c < 128) v = bl_tt[c - 64];
        else if (c < 192) v = br_tt[c - 128];
        else if (c < 224) v = bias_dt[c - 192];
        else              v = bias_tt[c - 224];
        bias_cat[c] = v;
    }
    for (int i = tid; i < 8 * 64; i += nthr) {
        const int r = i >> 6, c = i & 63;
        Wd[i] = (r < 5) ? Wl_dt[r * 64 + c] : 0.0f;
    }
    for (int c = tid; c < 64; c += nthr) bias_d[c] = bl_dt[c];
}

// ---------------------------------------------------------------------------
// Init per-call scratch: amax = -inf, den/num = 0 (must be idempotent).
// ---------------------------------------------------------------------------
__global__ void init_kernel(float* __restrict__ amax_dt, float* __restrict__ den_dt,
                            float* __restrict__ num_dt,
                            float* __restrict__ amax_tt, float* __restrict__ den_tt,
                            float* __restrict__ num_tt) {
    const int tid = blockIdx.x * blockDim.x + threadIdx.x;
    const int nthr = blockDim.x * gridDim.x;
    const float ninf = -__builtin_inff();
    for (int i = tid; i < NTASK * 2; i += nthr) {
        amax_dt[i] = ninf; amax_tt[i] = ninf;
        den_dt[i] = 0.0f;  den_tt[i] = 0.0f;
    }
    for (int i = tid; i < NTASK * 64; i += nthr) {
        num_dt[i] = 0.0f; num_tt[i] = 0.0f;
    }
}

// ---------------------------------------------------------------------------
// WMMA f32 16x16x4 GEMM: Y[M,256] = X[M,12] @ W[12,256] + bias.
// One wave per 16x16 output tile; 4 waves/block cover 4 N-tiles.
// ---------------------------------------------------------------------------
__global__ __launch_bounds__(128)
void gemm_tasks_kernel(const float* __restrict__ X, const float* __restrict__ W,
                       const float* __restrict__ bias, float* __restrict__ Y) {
    const int wave = threadIdx.x >> 5;
    const int lane = threadIdx.x & 31;
    const int m0 = blockIdx.x * 16;
    const int n0 = (blockIdx.y * 4 + wave) * 16;
    const int lm = lane & 15, lg = lane >> 4;
    v8f acc = {};
#pragma unroll
    for (int k = 0; k < 3; ++k) {       // K=12 as 3 steps of 4
        const int ka = k * 4 + lg * 2;  // A/B layout: K = vgpr + 2*lanegroup
        v2f a, b;
        a.x = X[(size_t)(m0 + lm) * 12 + ka];
        a.y = X[(size_t)(m0 + lm) * 12 + ka + 1];
        b.x = W[(size_t)ka * 256 + n0 + lm];
        b.y = W[(size_t)(ka + 1) * 256 + n0 + lm];
        acc = __builtin_amdgcn_wmma_f32_16x16x4_f32(false, a, false, b,
                                                    (short)0, acc, false, false);
    }
    const float bv = bias[n0 + lm];
#pragma unroll
    for (int v = 0; v < 8; ++v)         // D layout: M = v + 8*lanegroup, N = lane&15
        Y[(size_t)(m0 + v + 8 * lg) * 256 + n0 + lm] = acc[v] + bv;
}

// Y[M,64] = X[M,5] @ Wd[8,64] + bias  (K padded 5->8, 2 WMMA k-steps)
__global__ __launch_bounds__(128)
void gemm_data_kernel(const float* __restrict__ X, const float* __restrict__ W,
                      const float* __restrict__ bias, float* __restrict__ Y) {
    const int wave = threadIdx.x >> 5;
    const int lane = threadIdx.x & 31;
    const int m0 = blockIdx.x * 16;
    const int n0 = wave * 16;
    const int lm = lane & 15, lg = lane >> 4;
    v8f acc = {};
#pragma unroll
    for (int k = 0; k < 2; ++k) {
        const int ka = k * 4 + lg * 2;
        v2f a, b;
        a.x = (ka     < 5) ? X[(size_t)(m0 + lm) * 5 + ka]     : 0.0f;
        a.y = (ka + 1 < 5) ? X[(size_t)(m0 + lm) * 5 + ka + 1] : 0.0f;
        b.x = W[(size_t)ka * 64 + n0 + lm];
        b.y = W[(size_t)(ka + 1) * 64 + n0 + lm];
        acc = __builtin_amdgcn_wmma_f32_16x16x4_f32(false, a, false, b,
                                                    (short)0, acc, false, false);
    }
    const float bv = bias[n0 + lm];
#pragma unroll
    for (int v = 0; v < 8; ++v)
        Y[(size_t)(m0 + v + 8 * lg) * 64 + n0 + lm] = acc[v] + bv;
}

// ---------------------------------------------------------------------------
// Edge pass 1: alpha[e,h] = sum_c att[h,c]*leaky(xr[d]+xl[s](+eattr@We), 0.2)
// One wave32 per edge. Lane covers channels {2*lane, 2*lane+1}.
// Lanes 0-15 = head 0, lanes 16-31 = head 1; reduce inside 16-lane halves.
// ---------------------------------------------------------------------------
__global__ __launch_bounds__(256)
void edge_alpha_kernel(const int* __restrict__ src, const int* __restrict__ dst,
                       const float* __restrict__ xl, int ls,
                       const float* __restrict__ xr, int rs,
                       const float* __restrict__ eattr, const float* __restrict__ We,
                       const float* __restrict__ att,
                       float* __restrict__ alpha, float* __restrict__ amax, int E) {
    const int lane = threadIdx.x & 31;
    const int e = blockIdx.x * (blockDim.x >> 5) + (threadIdx.x >> 5);
    if (e >= E) return;
    const int s = src[e], d = dst[e];
    const int ch0 = lane << 1, ch1 = ch0 | 1;
    float z0 = xr[(size_t)d * rs + ch0] + xl[(size_t)s * ls + ch0];
    float z1 = xr[(size_t)d * rs + ch1] + xl[(size_t)s * ls + ch1];
    if (eattr) {  // edge-feature transform on the fly: [3] @ We[3,64]
        const float a0 = eattr[(size_t)e * 3 + 0];
        const float a1 = eattr[(size_t)e * 3 + 1];
        const float a2 = eattr[(size_t)e * 3 + 2];
        z0 += a0 * We[ch0] + a1 * We[64 + ch0] + a2 * We[128 + ch0];
        z1 += a0 * We[ch1] + a1 * We[64 + ch1] + a2 * We[128 + ch1];
    }
    z0 = (z0 > 0.0f) ? z0 : 0.2f * z0;
    z1 = (z1 > 0.0f) ? z1 : 0.2f * z1;
    float p = z0 * att[ch0] + z1 * att[ch1];
#pragma unroll
    for (int o = 8; o > 0; o >>= 1) p += __shfl_xor(p, o, 16);
    if ((lane & 15) == 0) {
        const int h = lane >> 4;
        alpha[(size_t)e * 2 + h] = p;
        atomic_max_float(&amax[(size_t)d * 2 + h], p);
    }
}

// ---------------------------------------------------------------------------
// Edge pass 2: ex = exp(alpha - amax[dst]); num[d,h,c] += ex*xl[s,h,c];
// den[d,h] += ex.  (normalization deferred to finalize)
// ---------------------------------------------------------------------------
__global__ __launch_bounds__(256)
void edge_agg_kernel(const int* __restrict__ src, const int* __restrict__ dst,
                     const float* __restrict__ xl, int ls,
                     const float* __restrict__ alpha, const float* __restrict__ amax,
                     float* __restrict__ num, float* __restrict__ den, int E) {
    const int lane = threadIdx.x & 31;
    const int e = blockIdx.x * (blockDim.x >> 5) + (threadIdx.x >> 5);
    if (e >= E) return;
    const int s = src[e], d = dst[e];
    const int h = lane >> 4;
    const float ex = __expf(alpha[(size_t)e * 2 + h] - amax[(size_t)d * 2 + h]);
    const int ch0 = lane << 1, ch1 = ch0 | 1;
    atomicAdd(&num[(size_t)d * 64 + ch0], ex * xl[(size_t)s * ls + ch0]);
    atomicAdd(&num[(size_t)d * 64 + ch1], ex * xl[(size_t)s * ls + ch1]);
    if ((lane & 15) == 0) atomicAdd(&den[(size_t)d * 2 + h], ex);
}

// ---------------------------------------------------------------------------
// Finalize: normalize per head, mean over heads, + residual, LayerNorm (C=32,
// one wave per node, lane = channel), leaky 0.01, concat [x | dt | tt].
// ---------------------------------------------------------------------------
__global__ __launch_bounds__(256)
void finalize_kernel(const float* __restrict__ tasks_x, const float* __restrict__ xt,
                     const float* __restrict__ num_dt, const float* __restrict__ den_dt,
                     const float* __restrict__ num_tt, const float* __restrict__ den_tt,
                     const float* __restrict__ ln1w, const float* __restrict__ ln1b,
                     const float* __restrict__ ln2w, const float* __restrict__ ln2b,
                     float* __restrict__ out) {
    const int lane = threadIdx.x & 31;
    const int t = blockIdx.x * (blockDim.x >> 5) + (threadIdx.x >> 5);
    if (t >= NTASK) return;

    float v_dt = 0.5f * (num_dt[(size_t)t * 64 + lane]      / (den_dt[(size_t)t * 2 + 0] + 1e-16f)
                       + num_dt[(size_t)t * 64 + 32 + lane] / (den_dt[(size_t)t * 2 + 1] + 1e-16f))
               + xt[(size_t)t * 256 + 192 + lane];   // res_dt = x@Wres_dt + bias_dt
    float v_tt = 0.5f * (num_tt[(size_t)t * 64 + lane]      / (den_tt[(size_t)t * 2 + 0] + 1e-16f)
                       + num_tt[(size_t)t * 64 + 32 + lane] / (den_tt[(size_t)t * 2 + 1] + 1e-16f))
               + xt[(size_t)t * 256 + 224 + lane];   // res_tt

    float outv[2];
    const float* lw[2] = { ln1w, ln2w };
    const float* lb[2] = { ln1b, ln2b };
    float xv[2] = { v_dt, v_tt };
#pragma unroll
    for (int i = 0; i < 2; ++i) {
        float m = xv[i];
#pragma unroll
        for (int o = 16; o > 0; o >>= 1) m += __shfl_xor(m, o, 32);
        m *= (1.0f / 32.0f);
        const float dv = xv[i] - m;
        float var = dv * dv;
#pragma unroll
        for (int o = 16; o > 0; o >>= 1) var += __shfl_xor(var, o, 32);
        var *= (1.0f / 32.0f);
        float y = dv * rsqrtf(var + 1e-5f) * lw[i][lane] + lb[i][lane];
        outv[i] = (y > 0.0f) ? y : 0.01f * y;
    }

    float* o = out + (size_t)t * 76;
    if (lane < 12) o[lane] = tasks_x[(size_t)t * 12 + lane];
    o[12 + lane] = outv[0];
    o[44 + lane] = outv[1];
}

// ---------------------------------------------------------------------------
extern "C" void kernel_launch(void* const* d_in, const int* in_sizes, int n_in,
                              void* d_out, int out_size, void* d_ws, size_t ws_size,
                              hipStream_t stream) {
    const float* tasks_x = (const float*)d_in[0];
    const float* data_x  = (const float*)d_in[1];
    const int*   ei_dt   = (const int*)d_in[2];
    const float* ea_dt   = (const float*)d_in[3];
    const int*   ei_tt   = (const int*)d_in[4];
    const float* Wl_dt   = (const float*)d_in[5];
    const float* bl_dt   = (const float*)d_in[6];
    const float* Wr_dt   = (const float*)d_in[7];
    const float* br_dt   = (const float*)d_in[8];
    const float* We_dt   = (const float*)d_in[9];
    const float* att_dt  = (const float*)d_in[10];
    const float* Wres_dt = (const float*)d_in[11];
    const float* bias_dt = (const float*)d_in[12];
    const float* Wl_tt   = (const float*)d_in[13];
    const float* bl_tt   = (const float*)d_in[14];
    const float* Wr_tt   = (const float*)d_in[15];
    const float* br_tt   = (const float*)d_in[16];
    const float* att_tt  = (const float*)d_in[17];
    const float* Wres_tt = (const float*)d_in[18];
    const float* bias_tt = (const float*)d_in[19];
    const float* ln1w    = (const float*)d_in[20];
    const float* ln1b    = (const float*)d_in[21];
    const float* ln2w    = (const float*)d_in[22];
    const float* ln2b    = (const float*)d_in[23];

    // workspace carve (floats)
    float* ws = (float*)d_ws;
    size_t off = 0;
    float* Wcat     = ws + off; off += 12 * 256;
    float* bias_cat = ws + off; off += 256;
    float* Wd       = ws + off; off += 8 * 64;
    float* bias_d   = ws + off; off += 64;
    float* xt       = ws + off; off += (size_t)NTASK * 256;  // [xr_dt|xl_tt|xr_tt|res_dt|res_tt]
    float* xld      = ws + off; off += (size_t)NDATA * 64;
    float* amax_dt  = ws + off; off += (size_t)NTASK * 2;
    float* den_dt   = ws + off; off += (size_t)NTASK * 2;
    float* num_dt   = ws + off; off += (size_t)NTASK * 64;
    float* alpha_dt = ws + off; off += (size_t)N_EDT * 2;
    float* amax_tt  = ws + off; off += (size_t)NTASK * 2;
    float* den_tt   = ws + off; off += (size_t)NTASK * 2;
    float* num_tt   = ws + off; off += (size_t)NTASK * 64;
    float* alpha_tt = ws + off; off += (size_t)N_ETT * 2;
    (void)ws_size; (void)n_in; (void)in_sizes; (void)out_size;

    prep_kernel<<<8, 256, 0, stream>>>(Wl_dt, bl_dt, Wr_dt, br_dt, Wres_dt, bias_dt,
                                       Wl_tt, bl_tt, Wr_tt, br_tt, Wres_tt, bias_tt,
                                       Wcat, bias_cat, Wd, bias_d);
    init_kernel<<<4096, 256, 0, stream>>>(amax_dt, den_dt, num_dt, amax_tt, den_tt, num_tt);

    // dense transforms on WMMA pipe
    gemm_tasks_kernel<<<dim3(NTASK / 16, 4), 128, 0, stream>>>(tasks_x, Wcat, bias_cat, xt);
    gemm_data_kernel<<<dim3(NDATA / 16, 1), 128, 0, stream>>>(data_x, Wd, bias_d, xld);

    const int* src_dt = ei_dt;            const int* dst_dt = ei_dt + N_EDT;
    const int* src_tt = ei_tt;            const int* dst_tt = ei_tt + N_ETT;

    // edge pass 1: attention logits + per-dst max
    edge_alpha_kernel<<<(N_EDT + 7) / 8, 256, 0, stream>>>(
        src_dt, dst_dt, xld, 64, xt + 0, 256, ea_dt, We_dt, att_dt, alpha_dt, amax_dt, N_EDT);
    edge_alpha_kernel<<<(N_ETT + 7) / 8, 256, 0, stream>>>(
        src_tt, dst_tt, xt + 64, 256, xt + 128, 256, nullptr, nullptr, att_tt, alpha_tt, amax_tt, N_ETT);

    // edge pass 2: exp + unnormalized weighted aggregation
    edge_agg_kernel<<<(N_EDT + 7) / 8, 256, 0, stream>>>(
        src_dt, dst_dt, xld, 64, alpha_dt, amax_dt, num_dt, den_dt, N_EDT);
    edge_agg_kernel<<<(N_ETT + 7) / 8, 256, 0, stream>>>(
        src_tt, dst_tt, xt + 64, 256, alpha_tt, amax_tt, num_tt, den_tt, N_ETT);

    // per-node epilogue
    finalize_kernel<<<(NTASK + 7) / 8, 256, 0, stream>>>(
        tasks_x, xt, num_dt, den_dt, num_tt, den_tt, ln1w, ln1b, ln2w, ln2b, (float*)d_out);
}